// SupraGNNClassifier_82154134438094
// MI455X (gfx1250) — compile-verified
//
#include <hip/hip_runtime.h>
#include <hip/hip_bf16.h>
#include <math.h>

#define N_NODES 100000
#define N_EDGES 1600000
#define N_GRAPHS 512
#define IN_DIM 256
#define HID 128
#define GOUT 64
#define SCAL 2

typedef __attribute__((ext_vector_type(2))) float v2f;
typedef __attribute__((ext_vector_type(8))) float v8f;

// ---------------- degree / dinv ----------------
__global__ void k_fill1(float* __restrict__ p, int n) {
  int i = blockIdx.x * blockDim.x + threadIdx.x;
  if (i < n) p[i] = 1.0f;  // self-loop contributes 1 to degree
}

__global__ void k_zero(float* __restrict__ p, int n) {
  int i = blockIdx.x * blockDim.x + threadIdx.x;
  if (i < n) p[i] = 0.0f;
}

__global__ void k_deg_accum(const int* __restrict__ dst, float* __restrict__ deg, int E) {
  int e = blockIdx.x * blockDim.x + threadIdx.x;
  if (e < E) atomicAdd(&deg[dst[e]], 1.0f);
}

__global__ void k_rsqrt(float* __restrict__ d, int n) {
  int i = blockIdx.x * blockDim.x + threadIdx.x;
  if (i < n) d[i] = rsqrtf(d[i]);
}

// ---------------- fp32 WMMA GEMM: C[M,N] = A[M,K] @ B[K,N] ----------------
// One 16-row slab per workgroup; wave w computes N-tile w. blockDim.x = (N/16)*32.
// Exact fp32 via V_WMMA_F32_16X16X4_F32 (K-step 4). M%16==0, N%16==0, K%4==0.
__global__ void k_gemm_wmma_f32(const float* __restrict__ A,
                                const float* __restrict__ B,
                                float* __restrict__ C,
                                int M, int K, int N) {
  const int lane = threadIdx.x & 31;
  const int wave = threadIdx.x >> 5;         // which N tile
  const int half = lane >> 4;                // 0: K lo pair, 1: K hi pair
  const int l    = lane & 15;
  const int tm   = blockIdx.x * 16;
  const int tn   = wave * 16;

  const float* Arow = A + (size_t)(tm + l) * K + 2 * half;
  const float* Bcol = B + (size_t)(2 * half) * N + tn + l;

  v8f acc = {};
  for (int k = 0; k < K; k += 4) {
    v2f a = *(const v2f*)(Arow + k);                 // A[tm+l][k+2h], A[tm+l][k+2h+1]
    v2f b;
    b.x = Bcol[(size_t)k * N];                       // B[k+2h][tn+l]
    b.y = Bcol[(size_t)k * N + N];                   // B[k+2h+1][tn+l]
    acc = __builtin_amdgcn_wmma_f32_16x16x4_f32(
        /*neg_a=*/false, a, /*neg_b=*/false, b,
        /*c_mod=*/(short)0, acc, /*reuse_a=*/false, /*reuse_b=*/false);
  }
  // D layout: VGPR r -> row tm + r + 8*half, col tn + l
  float* Crow = C + (size_t)(tm + 8 * half) * N + tn + l;
  #pragma unroll
  for (int r = 0; r < 8; ++r) Crow[(size_t)r * N] = acc[r];
}

// ---------------- GCN aggregation ----------------
// agg[n,:] = h[n,:] * dinv[n]^2   (self-loop term; also serves as init for atomics)
__global__ void k_self_init(const float* __restrict__ h, const float* __restrict__ dinv,
                            float* __restrict__ agg, int n, int F) {
  size_t i = (size_t)blockIdx.x * blockDim.x + threadIdx.x;
  size_t total = (size_t)n * F;
  if (i < total) {
    float d = dinv[i / F];
    agg[i] = h[i] * d * d;
  }
}

// one wave per edge: agg[dst,:] += h[src,:] * (dinv[src]*dinv[dst])
__global__ void k_edge_scatter(const float* __restrict__ h, const float* __restrict__ dinv,
                               const int* __restrict__ src, const int* __restrict__ dst,
                               float* __restrict__ agg, int E, int F) {
  int e = blockIdx.x * (blockDim.x >> 5) + (threadIdx.x >> 5);
  if (e >= E) return;
  int lane = threadIdx.x & 31;
  int s = src[e], d = dst[e];
  float norm = dinv[s] * dinv[d];
  const float* hs = h + (size_t)s * F;
  float* ad = agg + (size_t)d * F;
  for (int f = lane; f < F; f += 32)
    atomicAdd(&ad[f], hs[f] * norm);
}

__global__ void k_bias_relu(const float* __restrict__ agg, const float* __restrict__ b,
                            float* __restrict__ out, int n, int F) {
  size_t i = (size_t)blockIdx.x * blockDim.x + threadIdx.x;
  if (i < (size_t)n * F) out[i] = fmaxf(agg[i] + b[i % F], 0.0f);
}

// ---------------- pooling: pooled[g,:] += agg2[n,:]+b2 ; cnt[g] += 1 ----------------
__global__ void k_pool(const float* __restrict__ agg, const float* __restrict__ b,
                       const int* __restrict__ batch,
                       float* __restrict__ pooled, float* __restrict__ cnt, int n) {
  int node = blockIdx.x;
  int f = threadIdx.x;               // blockDim.x == GOUT
  if (node >= n) return;
  int g = batch[node];
  float v = agg[(size_t)node * GOUT + f] + b[f];
  atomicAdd(&pooled[(size_t)g * GOUT + f], v);
  if (f == 0) atomicAdd(&cnt[g], 1.0f);
}

// ---------------- MLP head: sigmoid(relu([emb,scal]@W3+b3)@W4+b4) ----------------
__global__ void k_head(const float* __restrict__ pooled, const float* __restrict__ cnt,
                       const float* __restrict__ scal,
                       const float* __restrict__ W3, const float* __restrict__ b3,
                       const float* __restrict__ W4, const float* __restrict__ b4,
                       float* __restrict__ out) {
  __shared__ float sh[HID];
  int g = blockIdx.x;
  int h = threadIdx.x;               // blockDim.x == HID
  float inv = 1.0f / fmaxf(cnt[g], 1.0f);
  float acc = b3[h];
  for (int i = 0; i < GOUT; ++i)
    acc += (pooled[(size_t)g * GOUT + i] * inv) * W3[i * HID + h];
  for (int i = 0; i < SCAL; ++i)
    acc += scal[g * SCAL + i] * W3[(GOUT + i) * HID + h];
  sh[h] = fmaxf(acc, 0.0f) * W4[h];
  __syncthreads();
  for (int s = HID / 2; s > 0; s >>= 1) {
    if (h < s) sh[h] += sh[h + s];
    __syncthreads();
  }
  if (h == 0) out[g] = 1.0f / (1.0f + expf(-(sh[0] + b4[0])));
}

extern "C" void kernel_launch(void* const* d_in, const int* in_sizes, int n_in,
                              void* d_out, int out_size, void* d_ws, size_t ws_size,
                              hipStream_t stream) {
  const float* x      = (const float*)d_in[0];
  const int*   ei     = (const int*)  d_in[1];   // [2, E] flat
  const int*   batch  = (const int*)  d_in[2];
  const float* scal   = (const float*)d_in[3];
  const float* W1     = (const float*)d_in[4];
  const float* b1     = (const float*)d_in[5];
  const float* W2     = (const float*)d_in[6];
  const float* b2     = (const float*)d_in[7];
  const float* W3     = (const float*)d_in[8];
  const float* b3     = (const float*)d_in[9];
  const float* W4     = (const float*)d_in[10];
  const float* b4     = (const float*)d_in[11];

  const int E = in_sizes[1] / 2;
  const int* src = ei;
  const int* dst = ei + E;

  // workspace layout (floats)
  float* ws     = (float*)d_ws;
  float* deg    = ws;                               // 100000 -> dinv in place
  float* pooled = ws + 100096;                      // 512*64
  float* cnt    = pooled + (size_t)N_GRAPHS * GOUT; // 512 (contiguous with pooled)
  float* bufA   = ws + 133376;                      // N*HID : h1, then relu(h1)
  float* bufB   = bufA + (size_t)N_NODES * HID;     // N*HID : agg1, then reused
  float* h2pre  = bufB;                             // N*GOUT (reuse)
  float* agg2   = bufB + (size_t)N_NODES * GOUT;    // N*GOUT (reuse)

  const int T = 256;
  float* out = (float*)d_out;

  // degree / dinv
  k_fill1    <<<(N_NODES + T - 1) / T, T, 0, stream>>>(deg, N_NODES);
  k_zero     <<<(N_GRAPHS * GOUT + N_GRAPHS + T - 1) / T, T, 0, stream>>>(pooled, N_GRAPHS * GOUT + N_GRAPHS);
  k_deg_accum<<<(E + T - 1) / T, T, 0, stream>>>(dst, deg, E);
  k_rsqrt    <<<(N_NODES + T - 1) / T, T, 0, stream>>>(deg, N_NODES);

  // conv1: h1 = x @ W1 ; agg1 = scatter + self ; bufA = relu(agg1 + b1)
  k_gemm_wmma_f32<<<N_NODES / 16, (HID / 16) * 32, 0, stream>>>(x, W1, bufA, N_NODES, IN_DIM, HID);
  {
    size_t tot = (size_t)N_NODES * HID;
    k_self_init<<<(tot + T - 1) / T, T, 0, stream>>>(bufA, deg, bufB, N_NODES, HID);
    k_edge_scatter<<<(E + 7) / 8, 256, 0, stream>>>(bufA, deg, src, dst, bufB, E, HID);
    k_bias_relu<<<(tot + T - 1) / T, T, 0, stream>>>(bufB, b1, bufA, N_NODES, HID);
  }

  // conv2: h2pre = relu_h1 @ W2 ; agg2 = scatter + self
  k_gemm_wmma_f32<<<N_NODES / 16, (GOUT / 16) * 32, 0, stream>>>(bufA, W2, h2pre, N_NODES, HID, GOUT);
  {
    size_t tot = (size_t)N_NODES * GOUT;
    k_self_init<<<(tot + T - 1) / T, T, 0, stream>>>(h2pre, deg, agg2, N_NODES, GOUT);
    k_edge_scatter<<<(E + 7) / 8, 256, 0, stream>>>(h2pre, deg, src, dst, agg2, E, GOUT);
  }

  // mean pool (+b2) and MLP head
  k_pool<<<N_NODES, GOUT, 0, stream>>>(agg2, b2, batch, pooled, cnt, N_NODES);
  k_head<<<N_GRAPHS, HID, 0, stream>>>(pooled, cnt, scal, W3, b3, W4, b4, out);
}